// Model_82231443849757
// MI455X (gfx1250) — compile-verified
//
#include <hip/hip_runtime.h>
#include <math.h>

// MI455X / gfx1250, wave32. Memory-bound workload (~580 MB traffic, <1 GFLOP):
// stream `sent` once for prototypes; map the two small GEMMs onto
// V_WMMA_F32_16X16X4_F32 (exact f32 — bytes-bound, not matrix-throughput-bound),
// and stage the 48 KB A-panels via the Tensor Data Mover (tensor_load_to_lds,
// TENSORcnt) with a manual-copy backstop.

typedef __attribute__((ext_vector_type(2))) float v2f;
typedef __attribute__((ext_vector_type(4))) float v4f;
typedef __attribute__((ext_vector_type(8))) float v8f;

#define N_CLS 128
#define K_SUP 5
#define DDIM  768
#define TTOK  1280
#define QTOK  256

// ---------------------------------------------------------------------------
// TDM: DMA a contiguous 16x768 f32 panel (48 KB) global -> LDS.
// D# built per CDNA5 ISA §8 (group0: count/lds_addr/global_addr/type=2;
// group1: data_size=4B, 1-row tile of 12288 elems). Groups 2/3 zero (<=2D).
// Issued by one wave; completion via s_wait_tensorcnt 0.
// This toolchain's builtin takes 6 args (g0 u32x4, g1 i32x8, g2 i32x4,
// g3 i32x4, g4 i32x8, cpol i32).
// ---------------------------------------------------------------------------
__device__ static inline void tdm_load_panel(const float* __restrict__ gsrc,
                                             float* lds_dst)
{
#if __has_builtin(__builtin_amdgcn_tensor_load_to_lds) && \
    __has_builtin(__builtin_amdgcn_s_wait_tensorcnt)
    typedef unsigned int u32x4 __attribute__((ext_vector_type(4)));
    typedef int          i32x8 __attribute__((ext_vector_type(8)));
    typedef int          i32x4 __attribute__((ext_vector_type(4)));

    const unsigned elems = 16u * DDIM;                       // 12288 f32
    unsigned long long ga = (unsigned long long)(uintptr_t)gsrc;
    unsigned lds = (unsigned)(uintptr_t)(void*)lds_dst;      // LDS byte offset

    u32x4 g0;
    g0[0] = 1u;                                              // count=1 (valid D#)
    g0[1] = lds;                                             // lds_addr [63:32]
    g0[2] = (unsigned)(ga & 0xFFFFFFFFu);                    // global_addr[31:0]
    g0[3] = (unsigned)((ga >> 32) & 0x01FFFFFFu)             // global_addr[56:32]
            | 0x80000000u;                                   // type=2 ("image")

    i32x8 g1;
    g1[0] = 0x00020000;                                      // data_size=2 (4B)
    g1[1] = (int)((elems & 0xFFFFu) << 16);                  // tensor_dim0[15:0]
    g1[2] = (int)(((elems >> 16) & 0xFFFFu) | (1u << 16));   // dim0 hi | tensor_dim1=1
    g1[3] = (int)((elems & 0xFFFFu) << 16);                  // tile_dim0 = 12288
    g1[4] = 1;                                               // tile_dim1 = 1
    g1[5] = (int)elems;                                      // tensor_dim0_stride
    g1[6] = 0;
    g1[7] = 0;

    i32x4 gz4 = {0, 0, 0, 0};
    i32x8 gz8 = {0, 0, 0, 0, 0, 0, 0, 0};
    __builtin_amdgcn_tensor_load_to_lds(g0, g1, gz4, gz4, gz8, 0);
    __builtin_amdgcn_s_wait_tensorcnt(0);
#else
    (void)gsrc; (void)lds_dst;
#endif
}

// ---------------------------------------------------------------------------
// Kernel 1: rel_prob / tri_prob — tiny 2-way softmax heads, block per n.
// ---------------------------------------------------------------------------
__global__ __launch_bounds__(256) void rel_tri_kernel(
    const float* __restrict__ s_pool, const float* __restrict__ tri_pool,
    const float* __restrict__ Wr, const float* __restrict__ br,
    const float* __restrict__ Wt, const float* __restrict__ bt,
    float* __restrict__ rel_out, float* __restrict__ tri_out)
{
    __shared__ v4f red[256];
    const int n = blockIdx.x;
    const int tid = threadIdx.x;
    float relAcc0 = 0.f, relAcc1 = 0.f, triAcc0 = 0.f, triAcc1 = 0.f;

    for (int k = 0; k < K_SUP; ++k) {
        float r0 = 0.f, r1 = 0.f, t0 = 0.f, t1 = 0.f;
        const float* sp = s_pool  + (size_t)(n * K_SUP + k) * DDIM;
        const float* tp = tri_pool + (size_t)(n * K_SUP + k) * DDIM;
        for (int d = tid; d < DDIM; d += 256) {
            float s = sp[d], t = tp[d];
            r0 += s * Wr[d];        r1 += s * Wr[DDIM + d];
            t0 += t * Wt[d];        t1 += t * Wt[DDIM + d];
        }
        v4f v; v.x = r0; v.y = r1; v.z = t0; v.w = t1;
        red[tid] = v;
        __syncthreads();
        for (int off = 128; off > 0; off >>= 1) {
            if (tid < off) red[tid] += red[tid + off];
            __syncthreads();
        }
        if (tid == 0) {
            v4f tot = red[0];
            float l0 = tot.x + br[0], l1 = tot.y + br[1];
            float m  = fmaxf(l0, l1);
            float e0 = __expf(l0 - m), e1 = __expf(l1 - m);
            float inv = 1.f / (e0 + e1);
            relAcc0 += e0 * inv;  relAcc1 += e1 * inv;
            l0 = tot.z + bt[0];  l1 = tot.w + bt[1];
            m  = fmaxf(l0, l1);
            e0 = __expf(l0 - m); e1 = __expf(l1 - m);
            inv = 1.f / (e0 + e1);
            triAcc0 += e0 * inv;  triAcc1 += e1 * inv;
        }
        __syncthreads();
    }
    if (tid == 0) {
        rel_out[n * 2 + 0] = relAcc0 * 0.2f;
        rel_out[n * 2 + 1] = relAcc1 * 0.2f;
        tri_out[n * 2 + 0] = triAcc0 * 0.2f;
        tri_out[n * 2 + 1] = triAcc1 * 0.2f;
    }
}

// ---------------------------------------------------------------------------
// Kernel 2: prototypes — streams sent[] (480 MB, the bandwidth hog).
// Grid: (n, d-tile of 256 cols); each wave load = 32 lanes x 4B = one full
// 128 B line, perfectly coalesced. Writes protos zero-padded to 16 rows per n.
// ---------------------------------------------------------------------------
__global__ __launch_bounds__(256) void proto_kernel(
    const float* __restrict__ sent, const int* __restrict__ tags,
    const int* __restrict__ valid, float* __restrict__ protosPad)
{
    __shared__ int s_tag[TTOK];
    __shared__ int s_val[TTOK];
    const int n     = blockIdx.x / 3;
    const int dtile = blockIdx.x % 3;
    const int tid   = threadIdx.x;
    const int d     = dtile * 256 + tid;

    for (int i = tid; i < TTOK; i += 256) {
        s_tag[i] = tags[n * TTOK + i];
        s_val[i] = valid[n * TTOK + i];
    }
    __syncthreads();

    float a0 = 0.f, a1 = 0.f, a2 = 0.f;
    int   c0 = 0,   c1 = 0,   c2 = 0;
    const float* base = sent + (size_t)n * TTOK * DDIM + d;
    for (int t = 0; t < TTOK; ++t) {
        float x = base[(size_t)t * DDIM];
        if (t + 16 < TTOK) __builtin_prefetch(&base[(size_t)(t + 16) * DDIM], 0, 1);
        int  tg = s_tag[t];
        bool v  = s_val[t] != 0;
        bool h0 = v && (tg == 0), h1 = v && (tg == 1), h2 = v && (tg == 2);
        a0 += h0 ? x : 0.f;  a1 += h1 ? x : 0.f;  a2 += h2 ? x : 0.f;
        c0 += h0;            c1 += h1;            c2 += h2;
    }

    float* P = protosPad + (size_t)n * 16 * DDIM + d;
    P[0 * DDIM] = a0 / fmaxf((float)c0, 1.f);
    P[1 * DDIM] = a1 / fmaxf((float)c1, 1.f);
    P[2 * DDIM] = a2 / fmaxf((float)c2, 1.f);
#pragma unroll
    for (int r = 3; r < 16; ++r) P[r * DDIM] = 0.f;   // zero padding rows
}

// ---------------------------------------------------------------------------
// Kernel 3: pw = protosPad[n] (16x768) @ W (768x768), f32 WMMA 16x16x4.
// A panel staged in LDS (TDM prefetch + manual-copy backstop), shared by the
// block's 8 waves; each wave owns one 16-wide column tile; 192 k-steps.
// ---------------------------------------------------------------------------
__global__ __launch_bounds__(256) void pw_kernel(
    const float* __restrict__ protosPad, const float* __restrict__ W,
    float* __restrict__ pwPad)
{
    __shared__ float Alds[16 * DDIM];            // 48 KB
    const int n     = blockIdx.x / 6;
    const int ctile = blockIdx.x % 6;
    const int tid   = threadIdx.x;

    const float* A = protosPad + (size_t)n * 16 * DDIM;

    if (tid < 32) tdm_load_panel(A, Alds);       // TDM DMA + s_wait_tensorcnt
    __syncthreads();                             // TDM writes settled
    for (int i = tid; i < 16 * DDIM; i += 256) Alds[i] = A[i];  // backstop
    __syncthreads();

    const int wave = tid >> 5;
    const int lane = tid & 31;
    const int cb   = ctile * 128 + wave * 16;    // output column base
    const int m    = lane & 15;                  // A row
    const int kh   = (lane >> 4) << 1;           // K sub-offset: 0 or 2
    const int nn   = lane & 15;                  // B/C column within tile

    v8f acc = {};
    for (int kk = 0; kk < DDIM; kk += 4) {
        const int ko = kk + kh;
        v2f a;                                   // A 16x4: lanes 0-15 K=0,1; 16-31 K=2,3
        a.x = Alds[m * DDIM + ko];
        a.y = Alds[m * DDIM + ko + 1];
        v2f b;                                   // B 4x16: rows striped across lanes
        b.x = W[(size_t)(ko)     * DDIM + cb + nn];
        b.y = W[(size_t)(ko + 1) * DDIM + cb + nn];
        acc = __builtin_amdgcn_wmma_f32_16x16x4_f32(
            false, a, false, b, (short)0, acc, false, false);
    }

    float* Out = pwPad + (size_t)n * 16 * DDIM;
    const int rbase = (lane >> 4) << 3;          // C/D: rows v (lanes<16) / v+8
#pragma unroll
    for (int v = 0; v < 8; ++v)
        Out[(size_t)(v + rbase) * DDIM + cb + nn] = acc[v];
}

// ---------------------------------------------------------------------------
// Kernel 4: ent_logits = pwPad[n] (16x768) @ q_sent[n]^T, then softmax over
// the 3 real prototype rows. B fragment is an 8B-aligned float2 per lane.
// ---------------------------------------------------------------------------
__global__ __launch_bounds__(256) void ent_kernel(
    const float* __restrict__ pwPad, const float* __restrict__ q_sent,
    float* __restrict__ ent_out)
{
    __shared__ float Alds[16 * DDIM];            // 48 KB
    const int n     = blockIdx.x >> 1;
    const int qtile = blockIdx.x & 1;
    const int tid   = threadIdx.x;

    const float* A = pwPad + (size_t)n * 16 * DDIM;

    if (tid < 32) tdm_load_panel(A, Alds);       // TDM DMA + s_wait_tensorcnt
    __syncthreads();
    for (int i = tid; i < 16 * DDIM; i += 256) Alds[i] = A[i];  // backstop
    __syncthreads();

    const int wave = tid >> 5;
    const int lane = tid & 31;
    const int qb   = qtile * 128 + wave * 16;
    const int m    = lane & 15;
    const int kh   = (lane >> 4) << 1;
    const int nn   = lane & 15;

    const float* Qrow = q_sent + (size_t)(n * QTOK + qb + nn) * DDIM;

    v8f acc = {};
    for (int kk = 0; kk < DDIM; kk += 4) {
        const int ko = kk + kh;
        v2f a;
        a.x = Alds[m * DDIM + ko];
        a.y = Alds[m * DDIM + ko + 1];
        v2f b = *(const v2f*)&Qrow[ko];          // K=ko, ko+1 for column qb+nn
        acc = __builtin_amdgcn_wmma_f32_16x16x4_f32(
            false, a, false, b, (short)0, acc, false, false);
    }

    // Rows 0..2 of the C tile live in acc[0..2] on lanes 0-15 (rows 8-15 on
    // lanes 16-31 are zero padding). Softmax over the 3 prototypes in-lane.
    if (lane < 16) {
        float l0 = acc[0], l1 = acc[1], l2 = acc[2];
        float mx = fmaxf(l0, fmaxf(l1, l2));
        float e0 = __expf(l0 - mx), e1 = __expf(l1 - mx), e2 = __expf(l2 - mx);
        float inv = 1.f / (e0 + e1 + e2);
        const int q = qb + nn;
        ent_out[((size_t)n * 3 + 0) * QTOK + q] = e0 * inv;
        ent_out[((size_t)n * 3 + 1) * QTOK + q] = e1 * inv;
        ent_out[((size_t)n * 3 + 2) * QTOK + q] = e2 * inv;
    }
}

// ---------------------------------------------------------------------------
extern "C" void kernel_launch(void* const* d_in, const int* in_sizes, int n_in,
                              void* d_out, int out_size, void* d_ws, size_t ws_size,
                              hipStream_t stream)
{
    const float* s_pool   = (const float*)d_in[0];
    const float* sent     = (const float*)d_in[1];
    const float* q_sent   = (const float*)d_in[2];
    const float* tri_pool = (const float*)d_in[3];
    const float* W        = (const float*)d_in[4];
    const float* Wr       = (const float*)d_in[5];
    const float* br       = (const float*)d_in[6];
    const float* Wt       = (const float*)d_in[7];
    const float* bt       = (const float*)d_in[8];
    const int*   tags     = (const int*)d_in[9];
    const int*   valid    = (const int*)d_in[10];

    float* out     = (float*)d_out;
    float* rel_out = out;                               // [128,2]
    float* ent_out = out + N_CLS * 2;                   // [128,3,256]
    float* tri_out = out + N_CLS * 2 + N_CLS * 3 * QTOK;

    float* protosPad = (float*)d_ws;                    // [128][16][768] f32
    float* pwPad     = protosPad + (size_t)N_CLS * 16 * DDIM;

    rel_tri_kernel<<<N_CLS,     256, 0, stream>>>(s_pool, tri_pool, Wr, br, Wt, bt,
                                                  rel_out, tri_out);
    proto_kernel  <<<N_CLS * 3, 256, 0, stream>>>(sent, tags, valid, protosPad);
    pw_kernel     <<<N_CLS * 6, 256, 0, stream>>>(protosPad, W, pwPad);
    ent_kernel    <<<N_CLS * 2, 256, 0, stream>>>(pwPad, q_sent, ent_out);
}